// GraphSAGE_85383949845212
// MI455X (gfx1250) — compile-verified
//
#include <hip/hip_runtime.h>

// GraphSAGE 2-layer forward for MI455X (gfx1250, wave32).
// f32 WMMA (V_WMMA_F32_16X16X4_F32) for the dense projections, L2-resident
// f32 atomic scatter for the mean aggregation, project-before-aggregate for
// layer 2 (linearity of mean-agg) to halve edge traffic.

typedef __attribute__((ext_vector_type(2))) float v2f;
typedef __attribute__((ext_vector_type(8))) float v8f;

#define F_IN  128
#define F_HID 128
#define F_OUT 64

__device__ __forceinline__ v8f wmma_f32_k4(v2f a, v2f b, v8f c) {
  // 8 args: (neg_a, A, neg_b, B, c_mod, C, reuse_a, reuse_b)
  return __builtin_amdgcn_wmma_f32_16x16x4_f32(false, a, false, b, (short)0, c,
                                               false, false);
}

__global__ void zero_kernel(float* __restrict__ p, long long n) {
  long long i = (long long)blockIdx.x * blockDim.x + threadIdx.x;
  if (i < n) p[i] = 0.0f;
}

// One 32-lane group per edge: lane0 bumps degree, each lane moves 4 floats
// (float4 load from x[src], 4 global f32 atomic adds into agg[dst]).
__global__ void edge_scatter1(const float* __restrict__ x,
                              const int* __restrict__ ei,
                              float* __restrict__ agg, float* __restrict__ deg,
                              int E) {
  long long t = (long long)blockIdx.x * blockDim.x + threadIdx.x;
  int e = (int)(t >> 5);
  int lane = (int)(t & 31);
  if (e >= E) return;
  int src = ei[e];
  int dst = ei[E + e];
  if (lane == 0) atomicAdd(&deg[dst], 1.0f);
  const float4* xs = (const float4*)(x + (long long)src * F_IN);
  float4 v = xs[lane];
  float* ad = agg + (long long)dst * F_IN + lane * 4;
  atomicAdd(ad + 0, v.x);
  atomicAdd(ad + 1, v.y);
  atomicAdd(ad + 2, v.z);
  atomicAdd(ad + 3, v.w);
}

__global__ void rdeg_kernel(const float* __restrict__ deg,
                            float* __restrict__ rdeg, int n) {
  int i = blockIdx.x * blockDim.x + threadIdx.x;
  if (i < n) rdeg[i] = 1.0f / fmaxf(deg[i], 1.0f);
}

// hpre = (agg * rdeg_row) @ Wl + bl + x @ Wr      [N x 128]
// 16-row output tile per block; 4 waves, each owning two 16-col tiles that are
// processed *inside* the k-loop with 4 independent accumulators (2 col tiles x
// 2 source GEMMs) so the XDL never stalls on an accumulator RAW chain.
// f32 WMMA 16x16x4 fragment layout (ISA 7.12.2):
//   A: lane l, row = l&15; lanes 0-15: {v0,v1} = K {0,1}; lanes 16-31: K {2,3}
//   B: col = l&15 per half-wave, same K split
//   C/D: VGPR i holds row i (lanes 0-15) / row i+8 (lanes 16-31), col = l&15
__global__ __launch_bounds__(128)
void gemm1_wmma(const float* __restrict__ x, const float* __restrict__ agg,
                const float* __restrict__ rdeg, const float* __restrict__ Wl,
                const float* __restrict__ bl, const float* __restrict__ Wr,
                float* __restrict__ out, int N_) {
  __shared__ float lA[16][F_IN];
  __shared__ float lX[16][F_IN];
  int rowBase = blockIdx.x * 16;
  int tid = threadIdx.x;
  for (int idx = tid; idx < 16 * F_IN; idx += 128) {
    int r = idx >> 7, c = idx & 127;
    int row = rowBase + r;
    if (row < N_) {
      long long g = (long long)row * F_IN + c;
      lA[r][c] = agg[g] * rdeg[row];
      lX[r][c] = x[g];
    } else {
      lA[r][c] = 0.0f;
      lX[r][c] = 0.0f;
    }
  }
  __syncthreads();

  int wave = tid >> 5, lane = tid & 31;
  int row = lane & 15, half = lane >> 4;
  int ln = lane & 15;
  int col0 = (wave * 2 + 0) * 16 + ln;
  int col1 = (wave * 2 + 1) * 16 + ln;

  v8f accL0 = {}, accR0 = {}, accL1 = {}, accR1 = {};
  for (int k = 0; k < F_IN; k += 4) {
    int kk = k + half * 2;
    v2f aA, aX, b0L, b0R, b1L, b1R;
    aA.x = lA[row][kk];
    aA.y = lA[row][kk + 1];
    aX.x = lX[row][kk];
    aX.y = lX[row][kk + 1];
    b0L.x = Wl[kk * F_HID + col0];
    b0L.y = Wl[(kk + 1) * F_HID + col0];
    b0R.x = Wr[kk * F_HID + col0];
    b0R.y = Wr[(kk + 1) * F_HID + col0];
    b1L.x = Wl[kk * F_HID + col1];
    b1L.y = Wl[(kk + 1) * F_HID + col1];
    b1R.x = Wr[kk * F_HID + col1];
    b1R.y = Wr[(kk + 1) * F_HID + col1];
    accL0 = wmma_f32_k4(aA, b0L, accL0);  // 4 independent chains
    accR0 = wmma_f32_k4(aX, b0R, accR0);
    accL1 = wmma_f32_k4(aA, b1L, accL1);
    accR1 = wmma_f32_k4(aX, b1R, accR1);
  }

  float bias0 = bl[(wave * 2 + 0) * 16 + ln];
  float bias1 = bl[(wave * 2 + 1) * 16 + ln];
  if (rowBase + 16 <= N_) {
    // Full tile: straight-line stores, no per-row exec-mask churn.
#pragma unroll
    for (int i = 0; i < 8; ++i) {
      long long m = rowBase + i + 8 * half;
      out[m * F_HID + col0] = accL0[i] + accR0[i] + bias0;
      out[m * F_HID + col1] = accL1[i] + accR1[i] + bias1;
    }
  } else {
    for (int i = 0; i < 8; ++i) {
      int m = rowBase + i + 8 * half;
      if (m < N_) {
        out[(long long)m * F_HID + col0] = accL0[i] + accR0[i] + bias0;
        out[(long long)m * F_HID + col1] = accL1[i] + accR1[i] + bias1;
      }
    }
  }
}

// Per-column partial sums / sums-of-squares over rows, reduced via L2 atomics.
__global__ void bn_partial(const float* __restrict__ h, float* __restrict__ gsum,
                           float* __restrict__ gsumsq, int N_) {
  int c = threadIdx.x;  // 0..127
  float s = 0.0f, ss = 0.0f;
  for (int r = blockIdx.x; r < N_; r += gridDim.x) {
    float v = h[(long long)r * F_HID + c];
    s += v;
    ss += v * v;
  }
  atomicAdd(&gsum[c], s);
  atomicAdd(&gsumsq[c], ss);
}

__global__ void bn_final(const float* __restrict__ gsum,
                         const float* __restrict__ gsumsq,
                         const float* __restrict__ gamma,
                         const float* __restrict__ beta,
                         float* __restrict__ scale, float* __restrict__ shift,
                         int N_) {
  int c = threadIdx.x;
  float inv_n = 1.0f / (float)N_;
  float mean = gsum[c] * inv_n;
  float var = gsumsq[c] * inv_n - mean * mean;  // biased var (train-mode)
  float sc = gamma[c] * rsqrtf(var + 1e-5f);
  scale[c] = sc;
  shift[c] = beta[c] - mean * sc;
}

__global__ void bn_apply_relu(float* __restrict__ h,
                              const float* __restrict__ scale,
                              const float* __restrict__ shift, long long n) {
  long long i = (long long)blockIdx.x * blockDim.x + threadIdx.x;
  if (i < n) {
    int c = (int)(i & (F_HID - 1));
    h[i] = fmaxf(h[i] * scale[c] + shift[c], 0.0f);
  }
}

// p = h @ Wl2 (projected-before-aggregation), out = h @ Wr2 + bl2.
// One 16-col tile per wave (F_OUT = 64 -> 4 tiles); shared A fragments feed
// two independent accumulator chains.
__global__ __launch_bounds__(128)
void gemm2_wmma(const float* __restrict__ h, const float* __restrict__ Wl,
                const float* __restrict__ bl, const float* __restrict__ Wr,
                float* __restrict__ p, float* __restrict__ out, int N_) {
  __shared__ float lH[16][F_HID];
  int rowBase = blockIdx.x * 16;
  int tid = threadIdx.x;
  for (int idx = tid; idx < 16 * F_HID; idx += 128) {
    int r = idx >> 7, c = idx & 127;
    int row = rowBase + r;
    lH[r][c] = (row < N_) ? h[(long long)row * F_HID + c] : 0.0f;
  }
  __syncthreads();

  int wave = tid >> 5, lane = tid & 31;
  int row = lane & 15, half = lane >> 4;
  int ln = lane & 15;
  int col = wave * 16 + ln;
  v8f accP = {}, accR = {};
  for (int k = 0; k < F_HID; k += 4) {
    int kk = k + half * 2;
    v2f a, bL, bR;
    a.x = lH[row][kk];
    a.y = lH[row][kk + 1];
    bL.x = Wl[kk * F_OUT + col];
    bL.y = Wl[(kk + 1) * F_OUT + col];
    bR.x = Wr[kk * F_OUT + col];
    bR.y = Wr[(kk + 1) * F_OUT + col];
    accP = wmma_f32_k4(a, bL, accP);
    accR = wmma_f32_k4(a, bR, accR);
  }
  float bias = bl[wave * 16 + ln];
  if (rowBase + 16 <= N_) {
#pragma unroll
    for (int i = 0; i < 8; ++i) {
      long long m = rowBase + i + 8 * half;
      p[m * F_OUT + col] = accP[i];
      out[m * F_OUT + col] = accR[i] + bias;
    }
  } else {
    for (int i = 0; i < 8; ++i) {
      int m = rowBase + i + 8 * half;
      if (m < N_) {
        p[(long long)m * F_OUT + col] = accP[i];
        out[(long long)m * F_OUT + col] = accR[i] + bias;
      }
    }
  }
}

// out[dst] += p[src] * rdeg[dst]; 16 lanes/edge, float4 loads from p.
__global__ void edge_scatter2(const float* __restrict__ p,
                              const int* __restrict__ ei,
                              const float* __restrict__ rdeg,
                              float* __restrict__ out, int E) {
  long long t = (long long)blockIdx.x * blockDim.x + threadIdx.x;
  int e = (int)(t >> 4);
  int lane = (int)(t & 15);
  if (e >= E) return;
  int src = ei[e];
  int dst = ei[E + e];
  float r = rdeg[dst];
  const float4* ps = (const float4*)(p + (long long)src * F_OUT);
  float4 v = ps[lane];
  float* od = out + (long long)dst * F_OUT + lane * 4;
  atomicAdd(od + 0, v.x * r);
  atomicAdd(od + 1, v.y * r);
  atomicAdd(od + 2, v.z * r);
  atomicAdd(od + 3, v.w * r);
}

extern "C" void kernel_launch(void* const* d_in, const int* in_sizes, int n_in,
                              void* d_out, int out_size, void* d_ws,
                              size_t ws_size, hipStream_t stream) {
  const float* x     = (const float*)d_in[0];
  const int*   ei    = (const int*)d_in[1];
  const float* Wl1   = (const float*)d_in[2];
  const float* bl1   = (const float*)d_in[3];
  const float* Wr1   = (const float*)d_in[4];
  const float* gamma = (const float*)d_in[5];
  const float* beta  = (const float*)d_in[6];
  const float* Wl2   = (const float*)d_in[7];
  const float* bl2   = (const float*)d_in[8];
  const float* Wr2   = (const float*)d_in[9];
  float* out = (float*)d_out;

  const int N_ = in_sizes[0] / F_IN;
  const int E  = in_sizes[1] / 2;

  // Workspace layout (floats). First span [deg | agg1 | gsum | gsumsq] is
  // contiguous so one zero kernel clears everything that needs zeroing.
  float* ws     = (float*)d_ws;
  float* deg    = ws;
  float* agg1   = deg + N_;
  float* gsum   = agg1 + (long long)N_ * F_IN;
  float* gsumsq = gsum + F_HID;
  float* rdeg   = gsumsq + F_HID;
  float* hpre   = rdeg + N_;
  float* p      = hpre + (long long)N_ * F_HID;
  float* scale  = p + (long long)N_ * F_OUT;
  float* shift  = scale + F_HID;
  (void)ws_size; (void)n_in; (void)out_size;

  long long zc = (long long)N_ + (long long)N_ * F_IN + 2 * F_HID;
  zero_kernel<<<(unsigned)((zc + 255) / 256), 256, 0, stream>>>(ws, zc);

  {
    long long t = (long long)E * 32;
    edge_scatter1<<<(unsigned)((t + 255) / 256), 256, 0, stream>>>(x, ei, agg1,
                                                                   deg, E);
  }
  rdeg_kernel<<<(N_ + 255) / 256, 256, 0, stream>>>(deg, rdeg, N_);

  gemm1_wmma<<<(N_ + 15) / 16, 128, 0, stream>>>(x, agg1, rdeg, Wl1, bl1, Wr1,
                                                 hpre, N_);

  bn_partial<<<256, 128, 0, stream>>>(hpre, gsum, gsumsq, N_);
  bn_final<<<1, 128, 0, stream>>>(gsum, gsumsq, gamma, beta, scale, shift, N_);
  {
    long long t = (long long)N_ * F_HID;
    bn_apply_relu<<<(unsigned)((t + 255) / 256), 256, 0, stream>>>(hpre, scale,
                                                                   shift, t);
  }

  gemm2_wmma<<<(N_ + 15) / 16, 128, 0, stream>>>(hpre, Wl2, bl2, Wr2, p, out,
                                                 N_);
  {
    long long t = (long long)E * 16;
    edge_scatter2<<<(unsigned)((t + 255) / 256), 256, 0, stream>>>(p, ei, rdeg,
                                                                   out, E);
  }
}